// DAGNLinkPrediction_26697516712280
// MI455X (gfx1250) — compile-verified
//
#include <hip/hip_runtime.h>
#include <hip/hip_bf16.h>

#define N_ENT 50000
#define N_REL 500
#define N_EDGE 500000
#define HEADS 2
#define DIM 64
#define HD 128           // HEADS*DIM
#define POW_ITER 3
#define ALPHA 0.15f
#define LN_EPS 1e-5f
#define LRELU 0.01f      // jax.nn.leaky_relu default slope

typedef __attribute__((ext_vector_type(2))) float    v2f;
typedef __attribute__((ext_vector_type(8))) float    v8f;
typedef __attribute__((ext_vector_type(4))) unsigned v4u;
typedef __attribute__((ext_vector_type(4))) int      v4i;
typedef __attribute__((ext_vector_type(8))) int      v8i;

#if defined(__gfx1250__) && __has_builtin(__builtin_amdgcn_tensor_load_to_lds)
#define HAVE_TDM 1
#else
#define HAVE_TDM 0
#endif

// ---- order-preserving float <-> uint mapping for atomicMax segment-max ----
static __device__ __forceinline__ unsigned fenc(float f) {
  unsigned u = __float_as_uint(f);
  return (u & 0x80000000u) ? ~u : (u | 0x80000000u);
}
static __device__ __forceinline__ float fdec(unsigned e) {
  unsigned u = (e & 0x80000000u) ? (e & 0x7FFFFFFFu) : ~e;
  return __uint_as_float(u);
}

// --------- Tensor Data Mover: stage a rows x cols f32 tensor into LDS ------
// D# built per CDNA5 ISA (2D tile == whole tensor).
static __device__ __forceinline__ void tdm_stage(float* lds_dst,
                                                 const float* gsrc,
                                                 int rows, int cols) {
#if HAVE_TDM
  if (threadIdx.x == 0) {
    unsigned long long ga = (unsigned long long)(uintptr_t)gsrc;
    unsigned lds_off = (unsigned)(uintptr_t)lds_dst;   // addr[31:0] = LDS offset
    v4u g0;
    g0.x = 0x1u;                                       // count=1, user mode
    g0.y = lds_off;                                    // lds_addr
    g0.z = (unsigned)(ga & 0xFFFFFFFFu);               // global_addr[31:0]
    g0.w = (unsigned)((ga >> 32) & 0x01FFFFFFu) | (2u << 30); // [120:96], type=2
    unsigned d0 = (unsigned)cols, d1 = (unsigned)rows;
    v8i g1;
    g1[0] = (int)(0x2u << 16);                 // wg_mask=0, data_size=2 (4B)
    g1[1] = (int)((d0 & 0xFFFFu) << 16);       // tensor_dim0[15:0]  @ bits 63:48
    g1[2] = (int)((d0 >> 16) | ((d1 & 0xFFFFu) << 16));   // dim0 hi | dim1 lo
    g1[3] = (int)((d1 >> 16) | ((d0 & 0xFFFFu) << 16));   // dim1 hi | tile_dim0
    g1[4] = (int)(d1 & 0xFFFFu);               // tile_dim1 | tile_dim2=0
    g1[5] = (int)d0;                           // tensor_dim0_stride[31:0]
    g1[6] = 0;                                 // stride0 hi | dim1_stride lo
    g1[7] = 0;
    v4i gz = {0, 0, 0, 0};                     // groups 2/3 unused (2D)
#if __clang_major__ >= 23
    v8i gz8 = {0, 0, 0, 0, 0, 0, 0, 0};
    __builtin_amdgcn_tensor_load_to_lds(g0, g1, gz, gz, gz8, 0);
#else
    __builtin_amdgcn_tensor_load_to_lds(g0, g1, gz, gz, 0);
#endif
    __builtin_amdgcn_s_wait_tensorcnt(0);
  }
#else
  for (int i = threadIdx.x; i < rows * cols; i += blockDim.x)
    lds_dst[i] = gsrc[i];
#endif
}

// ---------------------------------------------------------------- fills ----
__global__ void fill_f32(float* p, float v, int n) {
  int i = blockIdx.x * blockDim.x + threadIdx.x;
  if (i < n) p[i] = v;
}
__global__ void fill_u32(unsigned* p, unsigned v, int n) {
  int i = blockIdx.x * blockDim.x + threadIdx.x;
  if (i < n) p[i] = v;
}

// ------------------------------------------------------------ layernorm ----
// one wave32 per row of 64; lane holds elements lane and lane+32
__global__ void ln_kernel(const float* __restrict__ x, const float* __restrict__ g,
                          const float* __restrict__ b, float* __restrict__ out, int n) {
  int wave = threadIdx.x >> 5;
  int lane = threadIdx.x & 31;
  int row = blockIdx.x * 8 + wave;
  if (row >= n) return;
  float a0 = x[row * DIM + lane];
  float a1 = x[row * DIM + lane + 32];
  float s = a0 + a1;
  for (int m = 16; m >= 1; m >>= 1) s += __shfl_xor(s, m, 32);
  float mu = s * (1.0f / DIM);
  float d0 = a0 - mu, d1 = a1 - mu;
  float v = d0 * d0 + d1 * d1;
  for (int m = 16; m >= 1; m >>= 1) v += __shfl_xor(v, m, 32);
  float rs = rsqrtf(v * (1.0f / DIM) + LN_EPS);
  out[row * DIM + lane]      = d0 * rs * g[lane] + b[lane];
  out[row * DIM + lane + 32] = d1 * rs * g[lane + 32] + b[lane + 32];
}

// --------------------------------------------- node scores via f32 WMMA ----
// S[m,h] = sum_d tanh( (X @ W)[m, h*64+d] ) * att[h,d]
// 256 threads = 8 waves; block covers 16 rows x 128 cols; wave w owns 16-col
// tile. W staged to LDS by TDM; K loop = 16x V_WMMA_F32_16X16X4_F32.
__global__ void node_scores_wmma(const float* __restrict__ X,   // [M,64]
                                 const float* __restrict__ W,   // [64,128]
                                 const float* __restrict__ att, // [2,64]
                                 float* __restrict__ S,         // [M,2]
                                 int M) {
  __shared__ float sW[DIM * HD];      // 32 KB staged weights
  __shared__ float sred[16][HEADS];
  tdm_stage(sW, W, DIM, HD);
  if (threadIdx.x < 32) sred[threadIdx.x >> 1][threadIdx.x & 1] = 0.0f;
  __syncthreads();

  int lane = threadIdx.x & 31;
  int wave = threadIdx.x >> 5;          // 0..7 -> column tile
  int row_base = blockIdx.x * 16;
  int half  = lane >> 4;                // 0: K+0/1, 1: K+2/3 (A-layout 16x4 f32)
  int l16   = lane & 15;
  int a_row = min(row_base + l16, M - 1);   // clamp: no divergent loads
  int k_off = half * 2;
  int col   = wave * 16 + l16;          // 0..127
  const float2* X2 = reinterpret_cast<const float2*>(X);

  v8f c = {};
#pragma unroll
  for (int k0 = 0; k0 < DIM; k0 += 4) {
    float2 av = X2[(a_row * DIM + k0 + k_off) >> 1];   // global_load_b64
    v2f a, bm;
    a.x = av.x;
    a.y = av.y;
    bm.x = sW[(k0 + k_off) * HD + col];                // ds_load
    bm.y = sW[(k0 + k_off + 1) * HD + col];
    c = __builtin_amdgcn_wmma_f32_16x16x4_f32(false, a, false, bm, (short)0, c,
                                              false, false);
  }
  int head = col >> 6;                  // whole 16-col tile lies in one head
  int dim  = col & 63;
  float attv = att[head * DIM + dim];
  float acc[8];
#pragma unroll
  for (int j = 0; j < 8; ++j) {
    float v = tanhf(c[j]) * attv;       // c VGPR j holds rows j / j+8 (C layout)
    v += __shfl_xor(v, 1, 32);
    v += __shfl_xor(v, 2, 32);
    v += __shfl_xor(v, 4, 32);
    v += __shfl_xor(v, 8, 32);          // sum over the 16 cols of this tile
    acc[j] = v;
  }
  if (l16 == 0) {
#pragma unroll
    for (int j = 0; j < 8; ++j)
      atomicAdd(&sred[j + half * 8][head], acc[j]);    // ds_add_f32
  }
  __syncthreads();
  if (threadIdx.x < 32) {
    int r = threadIdx.x >> 1, hh = threadIdx.x & 1;
    int grow = row_base + r;
    if (grow < M) S[grow * HEADS + hh] = sred[r][hh];
  }
}

// -------------------------------------------------- edge softmax passes ----
__global__ void edge_max(const int* __restrict__ src, const int* __restrict__ dst,
                         const int* __restrict__ et, const float* __restrict__ sh,
                         const float* __restrict__ st, const float* __restrict__ sr,
                         unsigned* __restrict__ menc) {
  long gid = (long)blockIdx.x * blockDim.x + threadIdx.x;
  if (gid >= (long)N_EDGE * HEADS) return;
  int e = (int)(gid >> 1), h = (int)(gid & 1);
  int s = src[e], d = dst[e], r = et[e];
  float sc = sh[s * 2 + h] + st[d * 2 + h] + sr[r * 2 + h];
  sc = sc > 0.0f ? sc : LRELU * sc;
  atomicMax(&menc[s * 2 + h], fenc(sc));
}

__global__ void edge_exp(const int* __restrict__ src, const int* __restrict__ dst,
                         const int* __restrict__ et, const float* __restrict__ sh,
                         const float* __restrict__ st, const float* __restrict__ sr,
                         const unsigned* __restrict__ menc, float* __restrict__ Aw,
                         float* __restrict__ zsum) {
  long gid = (long)blockIdx.x * blockDim.x + threadIdx.x;
  if (gid >= (long)N_EDGE * HEADS) return;
  int e = (int)(gid >> 1), h = (int)(gid & 1);
  int s = src[e], d = dst[e], r = et[e];
  float sc = sh[s * 2 + h] + st[d * 2 + h] + sr[r * 2 + h];
  sc = sc > 0.0f ? sc : LRELU * sc;
  float ew = __expf(sc - fdec(menc[s * 2 + h]));
  Aw[e * 2 + h] = ew;
  atomicAdd(&zsum[s * 2 + h], ew);
}

__global__ void edge_norm(const int* __restrict__ src, float* __restrict__ Aw,
                          const float* __restrict__ zsum) {
  long gid = (long)blockIdx.x * blockDim.x + threadIdx.x;
  if (gid >= (long)N_EDGE * HEADS) return;
  int e = (int)(gid >> 1), h = (int)(gid & 1);
  Aw[e * 2 + h] = Aw[e * 2 + h] * (1.0f - ALPHA) / zsum[src[e] * 2 + h];
}

// -------------------------------------------------------- PPR iteration ----
__global__ void z_bcast(const float* __restrict__ h_ln, float* __restrict__ Z) {
  long gid = (long)blockIdx.x * blockDim.x + threadIdx.x;
  if (gid >= (long)N_ENT * HD) return;
  int n = (int)(gid >> 7), i = (int)(gid & 127);
  Z[gid] = h_ln[n * DIM + (i & 63)];
}
__global__ void z_init_alpha(const float* __restrict__ h_ln, float* __restrict__ Zn) {
  long gid = (long)blockIdx.x * blockDim.x + threadIdx.x;
  if (gid >= (long)N_ENT * HD) return;
  int n = (int)(gid >> 7), i = (int)(gid & 127);
  Zn[gid] = ALPHA * h_ln[n * DIM + (i & 63)];
}
// 128 consecutive threads per edge: coalesced gather of Z[dst] (25.6 MB, lives
// in the 192 MB L2), contiguous global_atomic_add_f32 scatter into Z_new[src].
__global__ void ppr_edge(const int* __restrict__ src, const int* __restrict__ dst,
                         const float* __restrict__ Aw, const float* __restrict__ Zc,
                         float* __restrict__ Zn) {
  long gid = (long)blockIdx.x * blockDim.x + threadIdx.x;
  if (gid >= (long)N_EDGE * HD) return;
  int e = (int)(gid >> 7), i = (int)(gid & 127), h = i >> 6;
  atomicAdd(&Zn[(long)src[e] * HD + i], Aw[e * 2 + h] * Zc[(long)dst[e] * HD + i]);
}

// --------------------------------------- output GEMM (f32 WMMA) + resid ----
// Out[N,64] = Xres + Z[N,128] @ Wo[128,64]; 128 threads = 4 waves, 16 rows per
// block; Wo staged to LDS by TDM; K loop = 32x V_WMMA_F32_16X16X4_F32.
__global__ void out_gemm_wmma(const float* __restrict__ Z, const float* __restrict__ Wo,
                              const float* __restrict__ Xres, float* __restrict__ Out) {
  __shared__ float sWo[HD * DIM];     // 32 KB staged weights
  tdm_stage(sWo, Wo, HD, DIM);
  __syncthreads();

  int lane = threadIdx.x & 31;
  int wave = threadIdx.x >> 5;          // 0..3 -> column tile
  int row_base = blockIdx.x * 16;
  int half = lane >> 4, l16 = lane & 15;
  int a_row = row_base + l16;
  int k_off = half * 2;
  int col = wave * 16 + l16;            // 0..63
  const float2* Z2 = reinterpret_cast<const float2*>(Z);

  v8f c = {};
#pragma unroll
  for (int k0 = 0; k0 < HD; k0 += 4) {
    float2 av = Z2[(a_row * HD + k0 + k_off) >> 1];    // global_load_b64
    v2f a, bm;
    a.x = av.x;
    a.y = av.y;
    bm.x = sWo[(k0 + k_off) * DIM + col];              // ds_load
    bm.y = sWo[(k0 + k_off + 1) * DIM + col];
    c = __builtin_amdgcn_wmma_f32_16x16x4_f32(false, a, false, bm, (short)0, c,
                                              false, false);
  }
#pragma unroll
  for (int j = 0; j < 8; ++j) {
    int row = row_base + j + half * 8;
    Out[row * DIM + col] = Xres[row * DIM + col] + c[j];
  }
}

// ------------------------------------------------------------- launcher ----
extern "C" void kernel_launch(void* const* d_in, const int* in_sizes, int n_in,
                              void* d_out, int out_size, void* d_ws, size_t ws_size,
                              hipStream_t stream) {
  (void)in_sizes; (void)n_in; (void)out_size; (void)ws_size;
  const float* ent0 = (const float*)d_in[0];
  const float* rel  = (const float*)d_in[1];
  const int*   eidx = (const int*)d_in[2];
  const int*   etyp = (const int*)d_in[3];
  const float* gam  = (const float*)d_in[4];
  const float* bet  = (const float*)d_in[5];
  const float* Wh   = (const float*)d_in[6];
  const float* Wt   = (const float*)d_in[7];
  const float* Wr   = (const float*)d_in[8];
  const float* ah   = (const float*)d_in[9];
  const float* at   = (const float*)d_in[10];
  const float* ar   = (const float*)d_in[11];
  const float* Wo   = (const float*)d_in[12];
  const int* esrc = eidx;
  const int* edst = eidx + N_EDGE;

  char* ws = (char*)d_ws;
  size_t off = 0;
  auto carve = [&](size_t bytes) -> char* {
    char* p = ws + off;
    off = (off + bytes + 255) & ~(size_t)255;
    return p;
  };
  float*    h_ln = (float*)carve((size_t)N_ENT * DIM * 4);
  float*    Za   = (float*)carve((size_t)N_ENT * HD * 4);
  float*    Zb   = (float*)carve((size_t)N_ENT * HD * 4);
  float*    ent  = (float*)carve((size_t)N_ENT * DIM * 4);
  float*    sh   = (float*)carve((size_t)N_ENT * HEADS * 4);
  float*    st   = (float*)carve((size_t)N_ENT * HEADS * 4);
  float*    sr   = (float*)carve((size_t)N_REL * HEADS * 4);
  unsigned* menc = (unsigned*)carve((size_t)N_ENT * HEADS * 4);
  float*    zsum = (float*)carve((size_t)N_ENT * HEADS * 4);
  float*    Aw   = (float*)carve((size_t)N_EDGE * HEADS * 4);

  hipMemcpyAsync(ent, ent0, (size_t)N_ENT * DIM * 4, hipMemcpyDeviceToDevice, stream);

  const int EB = 256;
  const int NH = N_ENT * HEADS;
  const long EH = (long)N_EDGE * HEADS;
  const int eg = (int)((EH + EB - 1) / EB);
  const long NZ = (long)N_ENT * HD;
  const int ng = (int)((NZ + EB - 1) / EB);
  const long EZ = (long)N_EDGE * HD;
  const int ezg = (int)((EZ + EB - 1) / EB);

  for (int l = 0; l < 2; ++l) {
    ln_kernel<<<N_ENT / 8, 256, 0, stream>>>(ent, gam + l * DIM, bet + l * DIM,
                                             h_ln, N_ENT);
    node_scores_wmma<<<N_ENT / 16, 256, 0, stream>>>(h_ln, Wh + l * DIM * HD,
                                                     ah + l * HEADS * DIM, sh, N_ENT);
    node_scores_wmma<<<N_ENT / 16, 256, 0, stream>>>(h_ln, Wt + l * DIM * HD,
                                                     at + l * HEADS * DIM, st, N_ENT);
    node_scores_wmma<<<(N_REL + 15) / 16, 256, 0, stream>>>(rel, Wr + l * DIM * HD,
                                                            ar + l * HEADS * DIM, sr,
                                                            N_REL);
    fill_u32<<<(NH + EB - 1) / EB, EB, 0, stream>>>(menc, 0u, NH);
    fill_f32<<<(NH + EB - 1) / EB, EB, 0, stream>>>(zsum, 0.0f, NH);
    edge_max<<<eg, EB, 0, stream>>>(esrc, edst, etyp, sh, st, sr, menc);
    edge_exp<<<eg, EB, 0, stream>>>(esrc, edst, etyp, sh, st, sr, menc, Aw, zsum);
    edge_norm<<<eg, EB, 0, stream>>>(esrc, Aw, zsum);

    float* Zc = Za;
    float* Zn = Zb;
    z_bcast<<<ng, EB, 0, stream>>>(h_ln, Zc);
    for (int it = 0; it < POW_ITER; ++it) {
      z_init_alpha<<<ng, EB, 0, stream>>>(h_ln, Zn);
      ppr_edge<<<ezg, EB, 0, stream>>>(esrc, edst, Aw, Zc, Zn);
      float* t = Zc; Zc = Zn; Zn = t;
    }
    float* dst_out = (l == 1) ? (float*)d_out : ent;
    out_gemm_wmma<<<N_ENT / 16, 128, 0, stream>>>(Zc, Wo + l * HD * DIM, ent, dst_out);
  }
}